// RayTracing_69836168233115
// MI455X (gfx1250) — compile-verified
//
#include <hip/hip_runtime.h>
#include <hip/hip_bf16.h>

// One thread per ray. All inner loops are select-based (no per-lane branches)
// so wave32 lanes never diverge inside the math; the only branch is a
// wave-uniform __ballot skip of the 100-sample sampler for waves whose rays
// all converged during sphere tracing.
//
// SDF identity: |cam + z*ray| - 1 == sqrt(z^2 + 2*(cam.ray)*z + |cam|^2) - 1
// (|ray| == 1), so every SDF eval is 1 FMA + 2 adds + 1 v_sqrt_f32.

#define THREADS       256
#define TRACE_ITERS_  10
#define N_STEPS_      100
#define N_ROOTFIND_   8
#define SDF_THR_      5e-05f
#define LS_STEP_      0.9f   /* (1 - LINE_SEARCH_STEP) / 2^0, LINE_STEP_ITERS == 1 */

__device__ __forceinline__ float sdf_at(float z, float two_b, float c2) {
    return sqrtf(fmaf(z, z + two_b, c2)) - 1.0f;
}

__global__ __launch_bounds__(THREADS)
void rt_sphere_trace_kernel(const float* __restrict__ cam_loc,
                            const float* __restrict__ ray_dir,
                            const int*   __restrict__ obj_mask,
                            float* __restrict__ out_pts,
                            float* __restrict__ out_obj,
                            float* __restrict__ out_dist,
                            int R, int N)
{
    __shared__ float s_ray[3 * THREADS];

    const int tid = threadIdx.x;
    const long long blk0 = (long long)blockIdx.x * THREADS;

    // ---- Stage this block's 256 ray vectors (768 dwords) into LDS via the
    // ---- CDNA5 async global->LDS path (ASYNCcnt), then consume from LDS.
    {
        const float* gbase = ray_dir + blk0 * 3;          // block-uniform -> SGPR pair
        // Flat LDS addresses carry the LDS byte offset in addr[31:0] (ISA 10.2).
        unsigned lds_base = (unsigned)(uintptr_t)(&s_ray[0]);
        long long lim = (long long)R * 3;
        #pragma unroll
        for (int i = 0; i < 3; ++i) {
            int e = i * THREADS + tid;                    // 0..767
            long long ge = blk0 * 3 + e;
            unsigned goff = (unsigned)(((ge < lim) ? e : 0) * 4);
            unsigned loff = lds_base + (unsigned)e * 4u;
            asm volatile("global_load_async_to_lds_b32 %0, %1, %2"
                         :: "v"(loff), "v"(goff), "s"(gbase) : "memory");
        }
        asm volatile("s_wait_asynccnt 0" ::: "memory");
        __syncthreads();
    }

    long long r  = blk0 + tid;
    bool valid   = (r < (long long)R);
    long long rc = valid ? r : (long long)(R - 1);

    const int bidx = (int)(rc / (long long)N);
    const float cx = cam_loc[bidx * 3 + 0];
    const float cy = cam_loc[bidx * 3 + 1];
    const float cz = cam_loc[bidx * 3 + 2];
    const float rx = s_ray[tid * 3 + 0];
    const float ry = s_ray[tid * 3 + 1];
    const float rz = s_ray[tid * 3 + 2];
    const int   obj = obj_mask[rc];

    const float bdot  = fmaf(rx, cx, fmaf(ry, cy, rz * cz));   // ray . cam
    const float c2    = fmaf(cx, cx, fmaf(cy, cy, cz * cz));   // |cam|^2
    const float two_b = 2.0f * bdot;

    // ---- Sphere intersection with the object-radius (R=5) sphere ----
    const float under = fmaf(bdot, bdot, 25.0f - c2);
    const bool  m0    = under > 0.0f;
    const float sq    = sqrtf(m0 ? under : 1.0f);
    const float near_ = m0 ? fmaxf(-sq - bdot, 0.0f) : 0.0f;
    const float far_  = m0 ? fmaxf( sq - bdot, 0.0f) : 0.0f;

    // ---- Sphere tracing (bidirectional, 10 iters + 1 line-search step) ----
    bool  unf_s = m0, unf_e = m0;
    float acc_s = near_, acc_e = far_;
    float next_s = unf_s ? sdf_at(acc_s, two_b, c2) : 0.0f;
    float next_e = unf_e ? sdf_at(acc_e, two_b, c2) : 0.0f;

    #pragma unroll
    for (int it = 0; it < TRACE_ITERS_; ++it) {
        float cur_s = unf_s ? next_s : 0.0f;  cur_s = (cur_s <= SDF_THR_) ? 0.0f : cur_s;
        float cur_e = unf_e ? next_e : 0.0f;  cur_e = (cur_e <= SDF_THR_) ? 0.0f : cur_e;
        unf_s = unf_s && (cur_s > SDF_THR_);
        unf_e = unf_e && (cur_e > SDF_THR_);
        acc_s += cur_s;
        acc_e -= cur_e;
        next_s = unf_s ? sdf_at(acc_s, two_b, c2) : 0.0f;
        next_e = unf_e ? sdf_at(acc_e, two_b, c2) : 0.0f;
        float cos_s = bdot + acc_s;          // ray . (cam + acc*ray)
        float cos_e = bdot + acc_e;
        bool np_s = (next_s < 0.0f) || (cos_s > 0.0f);
        bool np_e = (next_e < 0.0f) || (cos_e < 0.0f);
        // masked line-search back-off (LINE_STEP_ITERS == 1, step = 0.9)
        acc_s  = np_s ? fmaf(-LS_STEP_, cur_s, acc_s) : acc_s;
        acc_e  = np_e ? fmaf( LS_STEP_, cur_e, acc_e) : acc_e;
        next_s = np_s ? sdf_at(acc_s, two_b, c2) : next_s;
        next_e = np_e ? sdf_at(acc_e, two_b, c2) : next_e;
        bool open = acc_s < acc_e;
        unf_s = unf_s && open;
        unf_e = unf_e && open;
    }
    {   // final mask refresh (matches the torch while-loop break point)
        float cur = unf_s ? next_s : 0.0f;  cur = (cur <= SDF_THR_) ? 0.0f : cur;
        unf_s = unf_s && (cur > SDF_THR_);
    }

    bool  net_obj = acc_s < acc_e;
    float out_d   = acc_s;

    // ---- Ray sampler + bisection, only for waves with any unfinished ray ----
    if (__ballot(unf_s)) {
        const float smin = unf_s ? acc_s : 0.0f;
        const float smax = unf_s ? acc_e : 0.0f;
        const float dz   = smax - smin;
        const float inv  = 1.0f / (float)(N_STEPS_ - 1);

        float best_tmp = 3.4e38f;  int   idx   = 0;
        float z_idx    = smin;     float sdf_i = 0.0f;
        float best_sdf = 3.4e38f;  float z_out = smin;

        #pragma unroll 4
        for (int j = 0; j < N_STEPS_; ++j) {
            float t   = (float)j * inv;
            float z   = fmaf(t, dz, smin);
            float sv  = sdf_at(z, two_b, c2);
            float sgn = (sv > 0.0f) ? 1.0f : ((sv < 0.0f) ? -1.0f : 0.0f);
            float tmp = sgn * (float)(N_STEPS_ - j);
            bool b1 = tmp < best_tmp;    // first sign-flip sample (argmin, first hit)
            best_tmp = b1 ? tmp : best_tmp;
            idx      = b1 ? j   : idx;
            z_idx    = b1 ? z   : z_idx;
            sdf_i    = b1 ? sv  : sdf_i;
            bool b2 = sv < best_sdf;     // min-sdf sample (argmin, first hit)
            best_sdf = b2 ? sv : best_sdf;
            z_out    = b2 ? z  : z_out;
        }

        bool  net_surface = sdf_i < 0.0f;
        float samp_dist   = z_idx;
        bool  p_out = unf_s && !((obj != 0) && net_surface);
        samp_dist = p_out ? z_out : samp_dist;

        // bisection between last-positive and first-negative sample
        int   ilo = (idx > 0) ? (idx - 1) : 0;
        float lo  = fmaf((float)ilo * inv, dz, smin);
        float hi  = z_idx;
        #pragma unroll
        for (int k = 0; k < N_ROOTFIND_; ++k) {
            float mid  = 0.5f * (lo + hi);
            bool go_lo = sdf_at(mid, two_b, c2) > 0.0f;
            lo = go_lo ? mid : lo;
            hi = go_lo ? hi  : mid;
        }
        float z_root = 0.5f * (lo + hi);
        bool  sec    = net_surface && unf_s;
        samp_dist    = sec ? z_root : samp_dist;
        bool  s_nm   = unf_s && net_surface;

        out_d   = unf_s ? samp_dist : out_d;
        net_obj = unf_s ? s_nm      : net_obj;
    }

    // All output branches reduce to pts = cam + out_d * ray.
    const float px = fmaf(out_d, rx, cx);
    const float py = fmaf(out_d, ry, cy);
    const float pz = fmaf(out_d, rz, cz);

    if (valid) {
        out_pts[r * 3 + 0] = px;
        out_pts[r * 3 + 1] = py;
        out_pts[r * 3 + 2] = pz;
        out_obj[r]  = net_obj ? 1.0f : 0.0f;
        out_dist[r] = out_d;
    }
}

extern "C" void kernel_launch(void* const* d_in, const int* in_sizes, int n_in,
                              void* d_out, int out_size, void* d_ws, size_t ws_size,
                              hipStream_t stream) {
    const float* cam = (const float*)d_in[0];   // [B,3]
    const float* ray = (const float*)d_in[1];   // [B,N,3]
    const int*   obj = (const int*)d_in[2];     // [B,N]

    const int R = in_sizes[1] / 3;              // B*N rays
    const int B = in_sizes[0] / 3;
    const int N = R / (B > 0 ? B : 1);

    float* out      = (float*)d_out;
    float* out_pts  = out;                      // [R,3]
    float* out_obj  = out + (size_t)R * 3;      // [R]
    float* out_dist = out + (size_t)R * 4;      // [R]

    const int blocks = (R + THREADS - 1) / THREADS;
    rt_sphere_trace_kernel<<<blocks, THREADS, 0, stream>>>(
        cam, ray, obj, out_pts, out_obj, out_dist, R, N);
}